// CGRC_81183471829067
// MI455X (gfx1250) — compile-verified
//
#include <hip/hip_runtime.h>

#define N_USERS_C 200000
#define N_ITEMS_C 100000
#define NTOT_C    300000
#define EDGES_C   4800000
#define DIM_C     64

typedef __attribute__((ext_vector_type(2))) float v2f;
typedef __attribute__((ext_vector_type(8))) float v8f;

// ---------------------------------------------------------------------------
// x[0 : N_USERS*D] = user_w   (float4 vectorized copy)
// ---------------------------------------------------------------------------
__global__ void lgcn_copy_users(const float* __restrict__ uw, float* __restrict__ x) {
    int i = blockIdx.x * blockDim.x + threadIdx.x;
    const int n4 = N_USERS_C * DIM_C / 4;
    if (i < n4) ((float4*)x)[i] = ((const float4*)uw)[i];
}

// ---------------------------------------------------------------------------
// x_items[r, :] = item_w[r, :] + content[r, :] @ proj_w^T + proj_b
// fp32 WMMA: D(16x16) = A(16x4) * B(4x16) + C, 16 K-steps over D=64.
// One wave32 per 16-item tile; 4 N-tiles (64 cols) per wave.
// proj_w (64x64, 16KB) staged in LDS; B[k,n] = proj_w[n,k].
// ---------------------------------------------------------------------------
__global__ __launch_bounds__(256) void lgcn_proj_items(
        const float* __restrict__ content, const float* __restrict__ itemw,
        const float* __restrict__ projw,   const float* __restrict__ projb,
        float* __restrict__ xitems) {
    __shared__ float Wlds[DIM_C * DIM_C];            // proj_w row-major [n][k]
    for (int i = threadIdx.x; i < DIM_C * DIM_C / 4; i += blockDim.x)
        ((float4*)Wlds)[i] = ((const float4*)projw)[i];
    __syncthreads();

    const int wave = threadIdx.x >> 5;               // 0..7
    const int lane = threadIdx.x & 31;
    const int tile = blockIdx.x * 8 + wave;          // 16-row item tile
    if (tile * 16 >= N_ITEMS_C) return;              // wave-uniform; EXEC stays full

    // A layout (32-bit 16x4): lanes 0-15 -> M=lane, K=k0+{0,1}; lanes 16-31 -> K=k0+{2,3}
    const int m  = lane & 15;
    const int kp = (lane >> 4) << 1;                 // 0 or 2
    const float* arow = content + (size_t)(tile * 16 + m) * DIM_C;

    v8f acc0 = {}, acc1 = {}, acc2 = {}, acc3 = {};
#pragma unroll
    for (int k0 = 0; k0 < DIM_C; k0 += 4) {
        v2f a  = *(const v2f*)(arow + k0 + kp);
        // B layout (32-bit 4x16, mirror of A with N): lane n = m, K=k0+kp+{0,1}
        v2f b0 = *(const v2f*)(&Wlds[( 0 + m) * DIM_C + k0 + kp]);
        v2f b1 = *(const v2f*)(&Wlds[(16 + m) * DIM_C + k0 + kp]);
        v2f b2 = *(const v2f*)(&Wlds[(32 + m) * DIM_C + k0 + kp]);
        v2f b3 = *(const v2f*)(&Wlds[(48 + m) * DIM_C + k0 + kp]);
        acc0 = __builtin_amdgcn_wmma_f32_16x16x4_f32(false, a, false, b0, (short)0, acc0, false, false);
        acc1 = __builtin_amdgcn_wmma_f32_16x16x4_f32(false, a, false, b1, (short)0, acc1, false, false);
        acc2 = __builtin_amdgcn_wmma_f32_16x16x4_f32(false, a, false, b2, (short)0, acc2, false, false);
        acc3 = __builtin_amdgcn_wmma_f32_16x16x4_f32(false, a, false, b3, (short)0, acc3, false, false);
    }

    // D layout (32-bit 16x16): VGPR v -> row v (lanes 0-15) / row v+8 (lanes 16-31), col = lane&15
    const int rbase = tile * 16 + ((lane >> 4) << 3);
#pragma unroll
    for (int v = 0; v < 8; v++) {
        const int r = rbase + v;
        float* dst = xitems + (size_t)r * DIM_C;
        const float* iw = itemw + (size_t)r * DIM_C;
        dst[ 0 + m] = acc0[v] + iw[ 0 + m] + projb[ 0 + m];
        dst[16 + m] = acc1[v] + iw[16 + m] + projb[16 + m];
        dst[32 + m] = acc2[v] + iw[32 + m] + projb[32 + m];
        dst[48 + m] = acc3[v] + iw[48 + m] + projb[48 + m];
    }
}

// ---------------------------------------------------------------------------
// Elementwise helpers (float4 vectorized over N*D = 19.2M floats)
// ---------------------------------------------------------------------------
__global__ void lgcn_copy4(float* __restrict__ dst, const float* __restrict__ src, int n4) {
    int i = blockIdx.x * blockDim.x + threadIdx.x;
    if (i < n4) ((float4*)dst)[i] = ((const float4*)src)[i];
}

__global__ void lgcn_zero4(float* __restrict__ p, int n4) {
    int i = blockIdx.x * blockDim.x + threadIdx.x;
    if (i < n4) ((float4*)p)[i] = make_float4(0.f, 0.f, 0.f, 0.f);
}

__global__ void lgcn_acc_add(float* __restrict__ out, const float* __restrict__ y, int n4) {
    int i = blockIdx.x * blockDim.x + threadIdx.x;
    if (i < n4) {
        float4 o = ((float4*)out)[i];
        float4 v = ((const float4*)y)[i];
        o.x += v.x; o.y += v.y; o.z += v.z; o.w += v.w;
        ((float4*)out)[i] = o;
    }
}

__global__ void lgcn_acc_add_scale(float* __restrict__ out, const float* __restrict__ y, int n4) {
    int i = blockIdx.x * blockDim.x + threadIdx.x;
    if (i < n4) {
        float4 o = ((float4*)out)[i];
        float4 v = ((const float4*)y)[i];
        o.x = (o.x + v.x) * 0.25f; o.y = (o.y + v.y) * 0.25f;
        o.z = (o.z + v.z) * 0.25f; o.w = (o.w + v.w) * 0.25f;
        ((float4*)out)[i] = o;
    }
}

// ---------------------------------------------------------------------------
// COO SpMM scatter: y[row, :] += val * x[col, :]
// 16 threads per edge, float4 per thread -> coalesced 256B row segments.
// x/y are L2-resident (76.8 MB each vs 192 MB L2); native f32 atomics.
// ---------------------------------------------------------------------------
__global__ void lgcn_spmm(const int* __restrict__ rows, const int* __restrict__ cols,
                          const float* __restrict__ vals, const float* __restrict__ x,
                          float* __restrict__ y) {
    long long gid = (long long)blockIdx.x * blockDim.x + threadIdx.x;
    if (gid >= (long long)EDGES_C * 16) return;
    const int e = (int)(gid >> 4);
    const int c = ((int)gid & 15) << 2;
    const int r  = rows[e];
    const int cc = cols[e];
    const float v = vals[e];
    float4 xv = *(const float4*)(x + (size_t)cc * DIM_C + c);
    float* dst = y + (size_t)r * DIM_C + c;
#if defined(__HIP_DEVICE_COMPILE__)
    unsafeAtomicAdd(dst + 0, v * xv.x);
    unsafeAtomicAdd(dst + 1, v * xv.y);
    unsafeAtomicAdd(dst + 2, v * xv.z);
    unsafeAtomicAdd(dst + 3, v * xv.w);
#endif
}

// ---------------------------------------------------------------------------
extern "C" void kernel_launch(void* const* d_in, const int* in_sizes, int n_in,
                              void* d_out, int out_size, void* d_ws, size_t ws_size,
                              hipStream_t stream) {
    const int*   adj_rows = (const int*)d_in[0];
    const int*   adj_cols = (const int*)d_in[1];
    const float* adj_vals = (const float*)d_in[2];
    const float* content  = (const float*)d_in[3];
    const float* user_w   = (const float*)d_in[4];
    const float* item_w   = (const float*)d_in[5];
    const float* proj_w   = (const float*)d_in[6];
    const float* proj_b   = (const float*)d_in[7];
    float* out = (float*)d_out;

    float* x = (float*)d_ws;                         // [N, 64]
    float* y = x + (size_t)NTOT_C * DIM_C;           // [N, 64] ping-pong

    const int nd  = NTOT_C * DIM_C;                  // 19,200,000
    const int nd4 = nd / 4;                          // 4,800,000
    const int tpb = 256;

    // x0 = concat(user_w, item_w + content @ proj_w^T + proj_b)
    {
        int n4 = N_USERS_C * DIM_C / 4;
        lgcn_copy_users<<<(n4 + tpb - 1) / tpb, tpb, 0, stream>>>(user_w, x);
        int tiles  = N_ITEMS_C / 16;                 // 6250
        int blocks = (tiles + 7) / 8;                // 8 waves/block
        lgcn_proj_items<<<blocks, tpb, 0, stream>>>(content, item_w, proj_w, proj_b,
                                                    x + (size_t)N_USERS_C * DIM_C);
    }

    // acc = x0
    lgcn_copy4<<<(nd4 + tpb - 1) / tpb, tpb, 0, stream>>>(out, x, nd4);

    // 3 propagation layers, ping-pong x <-> y
    long long spmm_threads = (long long)EDGES_C * 16;
    int spmm_blocks = (int)((spmm_threads + tpb - 1) / tpb);
    for (int l = 0; l < 3; ++l) {
        lgcn_zero4<<<(nd4 + tpb - 1) / tpb, tpb, 0, stream>>>(y, nd4);
        lgcn_spmm<<<spmm_blocks, tpb, 0, stream>>>(adj_rows, adj_cols, adj_vals, x, y);
        if (l < 2)
            lgcn_acc_add<<<(nd4 + tpb - 1) / tpb, tpb, 0, stream>>>(out, y, nd4);
        else
            lgcn_acc_add_scale<<<(nd4 + tpb - 1) / tpb, tpb, 0, stream>>>(out, y, nd4);
        float* t = x; x = y; y = t;
    }
}